// GCNN_7112465842224
// MI455X (gfx1250) — compile-verified
//
#include <hip/hip_runtime.h>
#include <hip/hip_bf16.h>

// ---------------------------------------------------------------------------
// 2-layer GCN forward for MI455X (gfx1250, wave32).
//   GEMMs run on the WMMA pipe via V_WMMA_F32_16X16X4_F32 (fp32-exact).
//   Both WMMA operands are staged in LDS (A tile + transposed W panel) so the
//   inner loop is pure ds_load + v_wmma. SpMM (COO scatter) uses global f32
//   atomics; all intermediates (~120MB) are L2-resident (192MB L2).
// ---------------------------------------------------------------------------

#define GCN_N   20000
#define GCN_B   4
#define GCN_FIN 128
#define GCN_H   128
#define GCN_C   64
#define KDIM    128      // inner dim of both GEMMs
#define LDSPAD  4        // row padding (floats) -> conflict-free banks

typedef __attribute__((ext_vector_type(2))) float v2f;
typedef __attribute__((ext_vector_type(8))) float v8f;

// ---------------------------------------------------------------------------
// Fused batched GEMM: S[n, j] = sum_i A[n, b*a_boff + i] * W[b][i][o]
// where j in [0, B*ocols), b = j / ocols, o = j % ocols.
// Block = 4 waves; blockIdx.x -> 16-row tile, blockIdx.y -> group of 4
// 16-col tiles (all inside one b since ocols % 64 == 0).
//   LDS: As[16][128]   A tile (fused ReLU for layer 2), b128-staged
//        Wst[64][128]  W panel, stored TRANSPOSED so each lane's B fragment
//                      (two consecutive K values of one output column) is a
//                      single 8-byte, bank-conflict-free ds_load.
// Each wave then runs 32 back-to-back v_wmma_f32_16x16x4_f32 with the
// accumulator resident in its 8 C/D VGPRs.
// ---------------------------------------------------------------------------
__global__ __launch_bounds__(128)
void gcn_gemm_wmma_f32(const float* __restrict__ A, int a_ld, int a_boff, int relu_a,
                       const float* __restrict__ W, int ocols,
                       float* __restrict__ S, int s_ld)
{
    __shared__ float As [16][KDIM + LDSPAD];
    __shared__ float Wst[64][KDIM + LDSPAD];   // [out-col in group][k]

    const int wave = threadIdx.x >> 5;   // 0..3
    const int lane = threadIdx.x & 31;
    const int half = lane >> 4;          // 0: lanes 0-15, 1: lanes 16-31
    const int l16  = lane & 15;

    const int n0 = blockIdx.x * 16;      // row-tile base (N = 20000 = 1250*16)
    const int g  = blockIdx.y;           // group of 4 j-tiles (64 output cols)
    const int b  = (g * 64) / ocols;     // batch replica for this whole group
    const int og = (g * 64) - b * ocols; // group's first column inside W[b]

    // ---- cooperative A-tile load: 16 x 128, float4, fused ReLU ----
    const float* Ab = A + (size_t)n0 * a_ld + (size_t)b * a_boff;
    for (int idx = threadIdx.x; idx < 16 * (KDIM / 4); idx += 128) {
        int r = idx >> 5;                // / (KDIM/4)
        int q = idx & 31;                // quad within row
        float4 v = *(const float4*)(Ab + (size_t)r * a_ld + q * 4);
        if (relu_a) {
            v.x = fmaxf(v.x, 0.f); v.y = fmaxf(v.y, 0.f);
            v.z = fmaxf(v.z, 0.f); v.w = fmaxf(v.w, 0.f);
        }
        As[r][q * 4 + 0] = v.x; As[r][q * 4 + 1] = v.y;
        As[r][q * 4 + 2] = v.z; As[r][q * 4 + 3] = v.w;
    }

    // ---- cooperative W-panel load: 128 x 64, float4, stored transposed ----
    const float* Wg = W + (size_t)b * KDIM * ocols + og;
    for (int idx = threadIdx.x; idx < KDIM * (64 / 4); idx += 128) {
        int k = idx >> 4;                // / (64/4)
        int q = idx & 15;                // quad within the 64-wide row
        float4 v = *(const float4*)(Wg + (size_t)k * ocols + q * 4);
        Wst[q * 4 + 0][k] = v.x; Wst[q * 4 + 1][k] = v.y;
        Wst[q * 4 + 2][k] = v.z; Wst[q * 4 + 3][k] = v.w;
    }
    __syncthreads();

    const int col = wave * 16 + l16;     // this lane's output column in group

    // ---- K loop: 32 x v_wmma_f32_16x16x4_f32, acc lives in C/D VGPRs ----
    // A 16x4 layout: lanes 0-15 {K=0,1}, lanes 16-31 {K=2,3} (row = lane%16).
    // B 4x16 layout (symmetric): lanes 0-15 {K=0,1}, lanes 16-31 {K=2,3},
    // column = lane%16.
    v8f acc = {};
#pragma unroll
    for (int k = 0; k < KDIM; k += 4) {
        v2f av, bv;
        av.x = As[l16][k + 2 * half + 0];
        av.y = As[l16][k + 2 * half + 1];
        bv.x = Wst[col][k + 2 * half + 0];
        bv.y = Wst[col][k + 2 * half + 1];
        acc = __builtin_amdgcn_wmma_f32_16x16x4_f32(
                  false, av, false, bv, (short)0, acc, false, false);
    }

    // ---- store C/D: VGPR v -> row half*8+v, col = lane%16 ----
    float* Sp = S + (size_t)n0 * s_ld + g * 64 + wave * 16;
#pragma unroll
    for (int v = 0; v < 8; ++v)
        Sp[(size_t)(half * 8 + v) * s_ld + l16] = acc[v];
}

// ---------------------------------------------------------------------------
// COO SpMM scatter: Hout[rows[e], :] += vals[e] * S[cols[e], :]
// One block per edge; threads cover ldf features in float4 gathers,
// scattering with global_atomic_add_f32 (L2-resident destination).
// ---------------------------------------------------------------------------
__global__ void gcn_spmm_atomic(const int* __restrict__ rows,
                                const int* __restrict__ cols,
                                const float* __restrict__ vals,
                                const float* __restrict__ S,
                                float* __restrict__ Hout, int ldf)
{
    const int e = blockIdx.x;
    const int r = rows[e];
    const int c = cols[e];
    const float v = vals[e];
    const float4* src = (const float4*)(S + (size_t)c * ldf);
    float* dst = Hout + (size_t)r * ldf;
    const int nq = ldf >> 2;
    for (int q = threadIdx.x; q < nq; q += blockDim.x) {
        float4 s = src[q];
        int f = q * 4;
        atomicAdd(dst + f + 0, v * s.x);
        atomicAdd(dst + f + 1, v * s.y);
        atomicAdd(dst + f + 2, v * s.z);
        atomicAdd(dst + f + 3, v * s.w);
    }
}

// ---------------------------------------------------------------------------
__global__ void gcn_zero_f4(float4* __restrict__ p, long long n4)
{
    long long i = (long long)blockIdx.x * blockDim.x + threadIdx.x;
    if (i < n4) p[i] = make_float4(0.f, 0.f, 0.f, 0.f);
}

// ---------------------------------------------------------------------------
// Regression heads: y[b, n] = dot(h2[n, b*64 : b*64+64], reg_w[b]) + reg_b[b]
// ---------------------------------------------------------------------------
__global__ void gcn_regress(const float* __restrict__ h2,
                            const float* __restrict__ rw,
                            const float* __restrict__ rb,
                            float* __restrict__ y)
{
    int n = blockIdx.x * blockDim.x + threadIdx.x;
    if (n >= GCN_N) return;
    const float* h = h2 + (size_t)n * (GCN_B * GCN_C);
#pragma unroll
    for (int b = 0; b < GCN_B; ++b) {
        float acc = rb[b];
        const float* w = rw + b * GCN_C;
#pragma unroll 8
        for (int c = 0; c < GCN_C; ++c)
            acc += h[b * GCN_C + c] * w[c];
        y[(size_t)b * GCN_N + n] = acc;
    }
}

// ---------------------------------------------------------------------------
extern "C" void kernel_launch(void* const* d_in, const int* in_sizes, int n_in,
                              void* d_out, int out_size, void* d_ws, size_t ws_size,
                              hipStream_t stream)
{
    const float* x    = (const float*)d_in[0];   // [N, 128]
    const int*   rows = (const int*)  d_in[1];   // [E]
    const int*   cols = (const int*)  d_in[2];   // [E]
    const float* vals = (const float*)d_in[3];   // [E]
    const float* w0   = (const float*)d_in[4];   // [B, 128, 128]
    const float* w1   = (const float*)d_in[5];   // [B, 128, 64]
    const float* rw   = (const float*)d_in[6];   // [B, 64]
    const float* rb   = (const float*)d_in[7];   // [B]
    float* y = (float*)d_out;                    // [B, N]

    const int E = in_sizes[1];

    // Workspace ping-pong: bufA/bufB each N*512 f32 (41 MB).
    float* bufA = (float*)d_ws;                       // s1 [N,512], then s2 [N,256]
    float* bufB = bufA + (size_t)GCN_N * 512;         // h1 [N,512], then h2 [N,256]

    // ---- Layer 1 GEMM: s1 = x @ w0cat  (relu off, a_boff=0) ----
    {
        dim3 grid(GCN_N / 16, (GCN_B * GCN_H) / 64);  // (1250, 8)
        gcn_gemm_wmma_f32<<<grid, 128, 0, stream>>>(
            x, GCN_FIN, /*a_boff=*/0, /*relu=*/0, w0, GCN_H, bufA, GCN_B * GCN_H);
    }

    // ---- zero h1, SpMM1 scatter ----
    {
        long long n4 = (long long)GCN_N * 512 / 4;
        gcn_zero_f4<<<(unsigned)((n4 + 255) / 256), 256, 0, stream>>>((float4*)bufB, n4);
        gcn_spmm_atomic<<<E, 128, 0, stream>>>(rows, cols, vals, bufA, bufB, GCN_B * GCN_H);
    }

    // ---- Layer 2 GEMM: s2 = relu(h1) @ w1cat  (A-tile depends on b) ----
    {
        dim3 grid(GCN_N / 16, (GCN_B * GCN_C) / 64);  // (1250, 4)
        gcn_gemm_wmma_f32<<<grid, 128, 0, stream>>>(
            bufB, GCN_B * GCN_H, /*a_boff=*/GCN_H, /*relu=*/1, w1, GCN_C, bufA, GCN_B * GCN_C);
    }

    // ---- zero h2, SpMM2 scatter ----
    {
        long long n4 = (long long)GCN_N * 256 / 4;
        gcn_zero_f4<<<(unsigned)((n4 + 255) / 256), 256, 0, stream>>>((float4*)bufB, n4);
        gcn_spmm_atomic<<<E, 64, 0, stream>>>(rows, cols, vals, bufA, bufB, GCN_B * GCN_C);
    }

    // ---- regression heads ----
    gcn_regress<<<(GCN_N + 255) / 256, 256, 0, stream>>>(bufB, rw, rb, y);
}